// Timeseries_32143535243271
// MI455X (gfx1250) — compile-verified
//
#include <hip/hip_runtime.h>
#include <cstdint>

// Problem constants (match reference: T=1000, B=512, K=128)
#define T_STEPS 1000
#define B_ROWS  512
#define K_DIM   128
#define BK      (B_ROWS * K_DIM)   // elements per timestep slab
#define DEPTH   8                  // async prefetch depth (steps ahead)
#define WAVES_PER_WG 4
#define WG_SIZE (WAVES_PER_WG * 32)
#define NUM_WG  (B_ROWS / WAVES_PER_WG)

// ---- CDNA5 async global->LDS copy (ASYNCcnt-tracked), per-lane 16B ----
__device__ __forceinline__ void async_load_b128(uint32_t lds_off, const void* gaddr) {
    asm volatile("global_load_async_to_lds_b128 %0, %1, off"
                 :: "v"(lds_off), "v"(gaddr)
                 : "memory");
}
__device__ __forceinline__ void wait_async_le_12() {
    // 2 issues/step, DEPTH=8 -> 16 in flight. In-order completion: at iter t
    // (issued = 16+2t), asynccnt<=12 guarantees issues #1..#(2t+4) landed,
    // i.e. eps_t, perm_{t-1}  AND  eps_{t+1}, perm_t (for the pipelined fetch).
    asm volatile("s_wait_asynccnt 12" ::: "memory");
}

// Gather one element of the 128-wide row held as 4 dwords/lane (lane L owns
// k = 4L..4L+3). Source lane = pi>>2 (byte addr pi & 0x7C), component = pi & 3.
__device__ __forceinline__ float gather_one(int pi, int i0, int i1, int i2, int i3) {
    const int a  = pi & 0x7C;
    const int c0 = __builtin_amdgcn_ds_bpermute(a, i0);
    const int c1 = __builtin_amdgcn_ds_bpermute(a, i1);
    const int c2 = __builtin_amdgcn_ds_bpermute(a, i2);
    const int c3 = __builtin_amdgcn_ds_bpermute(a, i3);
    const bool b0 = (pi & 1) != 0;
    const bool b1 = (pi & 2) != 0;
    const int lo = b0 ? c1 : c0;
    const int hi = b0 ? c3 : c2;
    return __int_as_float(b1 ? hi : lo);
}

__global__ __launch_bounds__(WG_SIZE) void ts_scan_kernel(
    const float* __restrict__ init_state,  // [B,K]
    const float* __restrict__ eps,         // [T,B,K]
    const float* __restrict__ mult,        // [1]
    const float* __restrict__ scale,       // [1]
    const int*   __restrict__ perm,        // [T,K]
    float* __restrict__ out)               // [T,B,K]
{
    // Async staging only; the recurrence state lives in registers (bpermute gather).
    __shared__ __align__(16) float eps_stage[WAVES_PER_WG][DEPTH][K_DIM];
    __shared__ __align__(16) int   perm_stage[WAVES_PER_WG][DEPTH][K_DIM];

    const int tid  = threadIdx.x;
    const int w    = tid >> 5;
    const int lane = tid & 31;
    const int row  = blockIdx.x * WAVES_PER_WG + w;   // one wave == one row of B
    const int k0   = lane * 4;                        // lane owns k0..k0+3

    const float m = mult[0];
    const float s = scale[0];

    const float* eps_ptr  = eps  + (size_t)row * K_DIM + k0;  // +t*BK per step
    float*       out_ptr  = out  + (size_t)row * K_DIM + k0;
    const int*   perm_ptr = perm + k0;                        // +t*K per step

    const uint32_t lane_b    = (uint32_t)(lane * 16);
    const uint32_t eps_base  = (uint32_t)(uintptr_t)(&eps_stage[w][0][0]);
    const uint32_t perm_base = (uint32_t)(uintptr_t)(&perm_stage[w][0][0]);

    // ---- preloop: slot d <- (eps[d], perm[d-1]); slot 0's perm is unused.
    #pragma unroll
    for (int d = 0; d < DEPTH; ++d) {
        const int pd = (d == 0) ? 0 : d - 1;
        async_load_b128(eps_base  + (uint32_t)(d * K_DIM * 4) + lane_b,
                        eps_ptr  + (size_t)d * BK);
        async_load_b128(perm_base + (uint32_t)(d * K_DIM * 4) + lane_b,
                        perm_ptr + (size_t)pd * K_DIM);
    }

    const float4 p0v = *reinterpret_cast<const float4*>(init_state + (size_t)row * K_DIM + k0);

    float4 sp;      // samp_t for this lane's 4 elements (the register-resident state)
    float4 e_cur;   // eps for the *next* step (pipelined)
    int4   pi_cur;  // perm indices for the *next* step (pipelined)

    // ---- step 0 (peeled): samp0 = m*init + s*eps0 (identity gather)
    wait_async_le_12();
    {
        const float4 e0 = *reinterpret_cast<const float4*>(&eps_stage[w][0][k0]);
        sp.x = m * p0v.x + s * e0.x;
        sp.y = m * p0v.y + s * e0.y;
        sp.z = m * p0v.z + s * e0.z;
        sp.w = m * p0v.w + s * e0.w;
        *reinterpret_cast<float4*>(out_ptr) = sp;

        // pipelined fetch for step 1: eps_1 and perm[0] live in slot 1
        e_cur  = *reinterpret_cast<const float4*>(&eps_stage[w][1][k0]);
        pi_cur = *reinterpret_cast<const int4*>(&perm_stage[w][1][k0]);

        // refill slot 0: eps[8], perm[7]
        async_load_b128(eps_base  + lane_b, eps_ptr  + (size_t)DEPTH * BK);
        async_load_b128(perm_base + lane_b, perm_ptr + (size_t)(DEPTH - 1) * K_DIM);
    }

    // ---- main scan: samp_t = m * samp_{t-1}[perm_{t-1}] + s * eps_t
    for (int t = 1; t < T_STEPS; ++t) {
        wait_async_le_12();

        // critical chain: cross-lane gather of samp_{t-1} straight from registers
        const int i0 = __float_as_int(sp.x);
        const int i1 = __float_as_int(sp.y);
        const int i2 = __float_as_int(sp.z);
        const int i3 = __float_as_int(sp.w);
        const float g0 = gather_one(pi_cur.x, i0, i1, i2, i3);
        const float g1 = gather_one(pi_cur.y, i0, i1, i2, i3);
        const float g2 = gather_one(pi_cur.z, i0, i1, i2, i3);
        const float g3 = gather_one(pi_cur.w, i0, i1, i2, i3);

        sp.x = m * g0 + s * e_cur.x;
        sp.y = m * g1 + s * e_cur.y;
        sp.z = m * g2 + s * e_cur.z;
        sp.w = m * g3 + s * e_cur.w;

        *reinterpret_cast<float4*>(out_ptr + (size_t)t * BK) = sp;  // fire-and-forget

        // pipelined fetch for step t+1 (slot (t+1)&7 is guaranteed landed)
        const int slot1 = (t + 1) & (DEPTH - 1);
        e_cur  = *reinterpret_cast<const float4*>(&eps_stage[w][slot1][k0]);
        pi_cur = *reinterpret_cast<const int4*>(&perm_stage[w][slot1][k0]);

        // refill the slot consumed at this step (clamp at tail; overwrite harmless)
        const uint32_t slot_b = (uint32_t)(t & (DEPTH - 1)) * (K_DIM * 4);
        int tf = t + DEPTH;     if (tf > T_STEPS - 1) tf = T_STEPS - 1;
        int tp = t + DEPTH - 1; if (tp > T_STEPS - 1) tp = T_STEPS - 1;
        async_load_b128(eps_base  + slot_b + lane_b, eps_ptr  + (size_t)tf * BK);
        async_load_b128(perm_base + slot_b + lane_b, perm_ptr + (size_t)tp * K_DIM);
    }
}

extern "C" void kernel_launch(void* const* d_in, const int* in_sizes, int n_in,
                              void* d_out, int out_size, void* d_ws, size_t ws_size,
                              hipStream_t stream) {
    (void)in_sizes; (void)n_in; (void)out_size; (void)d_ws; (void)ws_size;
    const float* init  = (const float*)d_in[0];
    const float* eps   = (const float*)d_in[1];
    const float* mult  = (const float*)d_in[2];
    const float* scale = (const float*)d_in[3];
    const int*   perm  = (const int*)d_in[4];
    float* out = (float*)d_out;

    ts_scan_kernel<<<NUM_WG, WG_SIZE, 0, stream>>>(init, eps, mult, scale, perm, out);
}